// mHCResidual_48241072668958
// MI455X (gfx1250) — compile-verified
//
#include <hip/hip_runtime.h>
#include <math.h>

typedef float v2f __attribute__((ext_vector_type(2)));
typedef float v8f __attribute__((ext_vector_type(8)));

#define C_DIM     128
#define WAVES     4            // waves per block
#define ROWSTRIDE 132          // 128 + 4 pad -> conflict-free WMMA fragment reads
#define HSTRIDE   20           // 16 + 4 pad -> conflict-free transpose, float4 aligned
#define INV_SQRT512 0.04419417382415922f   // 1/sqrt(512)

// ---------------------------------------------------------------------------
// Kernel 1: fold the 4 replicated 128-column blocks of phi_res_w (16x512) and
// phi_post_w (4x512) into W~[o][c] = sum_j W[o][j*128+c].  fold layout in ws:
//   [0    .. 2047]  fold_res  (16 x 128)
//   [2048 .. 4095]  fold_post (16 x 128, rows 4..15 = 0)
// ---------------------------------------------------------------------------
__global__ void fold_weights_kernel(const float* __restrict__ res_w,
                                    const float* __restrict__ post_w,
                                    float* __restrict__ fold) {
    int idx = blockIdx.x * blockDim.x + threadIdx.x;
    if (idx >= 16 * 128) return;
    int n = idx >> 7, k = idx & 127;
    float s = 0.f;
#pragma unroll
    for (int j = 0; j < 4; ++j) s += res_w[n * 512 + j * 128 + k];
    fold[idx] = s;
    float sp = 0.f;
    if (n < 4) {
#pragma unroll
        for (int j = 0; j < 4; ++j) sp += post_w[n * 512 + j * 128 + k];
    }
    fold[2048 + idx] = sp;
}

// ---------------------------------------------------------------------------
// Main kernel: 4 waves/block, 16 positions per wave.
// ---------------------------------------------------------------------------
__launch_bounds__(WAVES * 32)
__global__ void mhc_main_kernel(const float* __restrict__ x,
                                const float* __restrict__ fo,
                                const float* __restrict__ fold,
                                const float* __restrict__ res_b,
                                const float* __restrict__ post_b,
                                const float* __restrict__ alpha_post_p,
                                const float* __restrict__ alpha_res_p,
                                float* __restrict__ out) {
    __shared__ float xs[WAVES][16][ROWSTRIDE];     // x tile per wave
    __shared__ float wres[16][ROWSTRIDE];          // folded res weights (B matrix)
    __shared__ float wpost[16][ROWSTRIDE];         // folded post weights (rows>=4 zero)
    __shared__ float sinv[WAVES][16];              // per-position inverse-norm scale
    __shared__ float hbuf[WAVES][16][HSTRIDE];     // transposed 4x4 gate matrices
    __shared__ float pbuf[WAVES][16][4];           // transposed post gates
    __shared__ float stot_res[WAVES][16];          // sum of doubly-stochastic 4x4
    __shared__ float stot_post[WAVES][16];         // sum of post gates

    const int tid  = threadIdx.x;
    const int wave = tid >> 5;
    const int lane = tid & 31;
    const int hi   = lane >> 4;       // half-wave select (K split / row split)
    const int l15  = lane & 15;

    const int tile_base = (blockIdx.x * WAVES + wave) * 16;

    // ---- stage folded weights (block-cooperative) ----
    for (int i = tid; i < 16 * 128; i += blockDim.x) {
        int n = i >> 7, k = i & 127;
        wres[n][k]  = fold[i];
        wpost[n][k] = fold[2048 + i];
    }

    // ---- stage x tile via CDNA5 async direct-to-LDS copies (ASYNCcnt path),
    //      512B coalesced per instruction; prefetch the f tile for the epilogue ----
    {
        const unsigned lds_base = (unsigned)(size_t)&xs[wave][0][lane * 4];
#pragma unroll
        for (int r = 0; r < 16; ++r) {
            const unsigned laddr = lds_base + (unsigned)(r * ROWSTRIDE * 4);
            const unsigned moff  = (unsigned)(((tile_base + r) * C_DIM + lane * 4) *
                                              (int)sizeof(float));
            asm volatile("global_load_async_to_lds_b128 %0, %1, %2"
                         :: "v"(laddr), "v"(moff), "s"(x)
                         : "memory");
            __builtin_prefetch(fo + (size_t)(tile_base + r) * C_DIM + lane * 4, 0, 3);
        }
        asm volatile("s_wait_asynccnt 0x0" ::: "memory");
    }
    __syncthreads();

    // ---- per-position inverse norm: x_flat is 4 copies -> ||x_flat|| = 2||x|| ----
    if (lane < 16) {
        float ss = 0.f;
#pragma unroll
        for (int c = 0; c < 128; c += 4) {
            float4 v = *(const float4*)&xs[wave][lane][c];
            ss += v.x * v.x + v.y * v.y + v.z * v.z + v.w * v.w;
        }
        float norm_flat = 2.0f * __builtin_sqrtf(ss);
        sinv[wave][lane] = __builtin_amdgcn_rcpf(norm_flat * INV_SQRT512 + 1e-12f);
    }
    __syncthreads();

    // ---- gate GEMV on the matrix unit: D(16 pos x 16 out) = A(16x4) x B(4x16), K-loop ----
    // A frag: lane holds row (lane&15), K = 4*kc + 2*hi + {0,1}
    // B frag: lane holds col (lane&15), same K split  (B[k][n] = W~[n][k])
    v8f acc_r = {};
    v8f acc_p = {};
#pragma unroll 4
    for (int kc = 0; kc < 32; ++kc) {
        const int col = kc * 4 + hi * 2;
        v2f a, br, bp;
        a.x  = xs[wave][l15][col];
        a.y  = xs[wave][l15][col + 1];
        br.x = wres[l15][col];
        br.y = wres[l15][col + 1];
        bp.x = wpost[l15][col];
        bp.y = wpost[l15][col + 1];
        acc_r = __builtin_amdgcn_wmma_f32_16x16x4_f32(false, a, false, br, (short)0,
                                                      acc_r, false, false);
        acc_p = __builtin_amdgcn_wmma_f32_16x16x4_f32(false, a, false, bp, (short)0,
                                                      acc_p, false, false);
    }

    const float a_post = *alpha_post_p;
    const float a_res  = *alpha_res_p;
    const float brb = res_b[l15];                  // bias for res output o = lane&15

    // D layout: VGPR j, lanes 0-15 -> position j, lanes 16-31 -> position j+8; col = lane&15
    float inv8[8];
#pragma unroll
    for (int j = 0; j < 8; ++j) inv8[j] = sinv[wave][j + 8 * hi];

    // ---- scale+bias+clip+exp, then transpose through LDS so each lane owns
    //      one position's full 4x4 Sinkhorn matrix in registers ----
#pragma unroll
    for (int j = 0; j < 8; ++j) {
        float v = a_res * (acc_r[j] * inv8[j] + brb);
        v = fminf(fmaxf(v, -10.f), 10.f);
        hbuf[wave][j + 8 * hi][l15] = __expf(v);
    }
    if (l15 < 4) {
        const float bpb = post_b[l15];
#pragma unroll
        for (int j = 0; j < 8; ++j) {
            float g = a_post * (acc_p[j] * inv8[j] + bpb);
            pbuf[wave][j + 8 * hi][l15] = 2.0f * __builtin_amdgcn_rcpf(1.0f + __expf(-g));
        }
    }
    __syncthreads();

    // ---- per-lane Sinkhorn-Knopp: lane handles position (lane&15); both
    //      half-waves compute redundantly (no divergence), hi==0 publishes ----
    float h[16];
    {
        float4 h0 = *(const float4*)&hbuf[wave][l15][0];
        float4 h1 = *(const float4*)&hbuf[wave][l15][4];
        float4 h2 = *(const float4*)&hbuf[wave][l15][8];
        float4 h3 = *(const float4*)&hbuf[wave][l15][12];
        h[0] = h0.x;  h[1] = h0.y;  h[2] = h0.z;  h[3] = h0.w;
        h[4] = h1.x;  h[5] = h1.y;  h[6] = h1.z;  h[7] = h1.w;
        h[8] = h2.x;  h[9] = h2.y;  h[10] = h2.z; h[11] = h2.w;
        h[12] = h3.x; h[13] = h3.y; h[14] = h3.z; h[15] = h3.w;
    }
#pragma unroll 2
    for (int it = 0; it < 20; ++it) {
#pragma unroll
        for (int r = 0; r < 4; ++r) {
            float s = ((h[4 * r] + h[4 * r + 1]) + (h[4 * r + 2] + h[4 * r + 3])) + 1e-12f;
            float rc = __builtin_amdgcn_rcpf(s);
            h[4 * r] *= rc; h[4 * r + 1] *= rc; h[4 * r + 2] *= rc; h[4 * r + 3] *= rc;
        }
#pragma unroll
        for (int c = 0; c < 4; ++c) {
            float s = ((h[c] + h[c + 4]) + (h[c + 8] + h[c + 12])) + 1e-12f;
            float rc = __builtin_amdgcn_rcpf(s);
            h[c] *= rc; h[c + 4] *= rc; h[c + 8] *= rc; h[c + 12] *= rc;
        }
    }
    float rt = (((h[0] + h[1]) + (h[2] + h[3])) + ((h[4] + h[5]) + (h[6] + h[7]))) +
               (((h[8] + h[9]) + (h[10] + h[11])) + ((h[12] + h[13]) + (h[14] + h[15])));
    float4 pv = *(const float4*)&pbuf[wave][l15][0];
    float pt = (pv.x + pv.y) + (pv.z + pv.w);
    if (hi == 0) {
        stot_res[wave][l15]  = rt;
        stot_post[wave][l15] = pt;
    }
    __syncthreads();

    // ---- output: out = x * sum(H_res) + sum(H_post) * f ; x reread from LDS ----
    const int c4 = lane * 4;
#pragma unroll 4
    for (int r = 0; r < 16; ++r) {
        const size_t p = (size_t)(tile_base + r);
        const float rt2 = stot_res[wave][r];
        const float pt2 = stot_post[wave][r];
        float4 xv = *(const float4*)&xs[wave][r][c4];
        float4 fv = *(const float4*)(fo + p * C_DIM + c4);
        float4 o;
        o.x = xv.x * rt2 + pt2 * fv.x;
        o.y = xv.y * rt2 + pt2 * fv.y;
        o.z = xv.z * rt2 + pt2 * fv.z;
        o.w = xv.w * rt2 + pt2 * fv.w;
        *(float4*)(out + p * C_DIM + c4) = o;
    }
}

// ---------------------------------------------------------------------------
extern "C" void kernel_launch(void* const* d_in, const int* in_sizes, int n_in,
                              void* d_out, int out_size, void* d_ws, size_t ws_size,
                              hipStream_t stream) {
    const float* x      = (const float*)d_in[0];
    const float* fo     = (const float*)d_in[1];
    // d_in[2]=phi_pre_w, d_in[3]=phi_pre_b : dead code in reference output
    const float* post_w = (const float*)d_in[4];
    const float* post_b = (const float*)d_in[5];
    const float* res_w  = (const float*)d_in[6];
    const float* res_b  = (const float*)d_in[7];
    // d_in[8]=alpha_pre unused
    const float* a_post = (const float*)d_in[9];
    const float* a_res  = (const float*)d_in[10];
    float* out  = (float*)d_out;
    float* fold = (float*)d_ws;      // needs 4096 floats = 16 KB

    const int P = in_sizes[0] / C_DIM;            // B*S positions

    fold_weights_kernel<<<(16 * 128 + 255) / 256, 256, 0, stream>>>(res_w, post_w, fold);

    const int blocks = P / (WAVES * 16);          // 16 positions per wave
    mhc_main_kernel<<<blocks, WAVES * 32, 0, stream>>>(x, fo, fold, res_b, post_b,
                                                       a_post, a_res, out);
}